// AffinityLoss_86517821214340
// MI455X (gfx1250) — compile-verified
//
#include <hip/hip_runtime.h>

// ---------------------------------------------------------------------------
// AffinityLoss on MI455X (gfx1250):
//   per-pixel 9x19 Gram matrix via V_WMMA_F32_16X16X4_F32. For a symmetric
//   Gram D = A*A^T, the ISA's B-fragment layout is bit-identical to the
//   A-fragment layout, so the same registers feed both operands.
//   BCE-with-logits via raw v_exp_f32/v_log_f32 (arguments are range-safe);
//   hierarchical deterministic reduction.
// ---------------------------------------------------------------------------

typedef __attribute__((ext_vector_type(2))) float v2f;
typedef __attribute__((ext_vector_type(8))) float v8f;

#define CC   19            // channels
#define HH   384
#define WW   384
#define WP   382           // w - k + 1
#define LPX  (382 * 382)   // 145924 windows per image
#define NPIX (4 * LPX)     // 583696 total windows
#define WAVES_PER_BLOCK 8
#define NBLOCKS (NPIX / WAVES_PER_BLOCK)   // 72962 exactly (no partial waves)
#define PLANE (HH * WW)    // 147456

// 1 / (n * 81 * L) = 1 / 47,279,376
#define INV_TOTAL (1.0f / 47279376.0f)

#define LOG2E 1.4426950408889634f
#define LN2   0.6931471805599453f

__device__ __forceinline__ float bce_term(float x, float y)
{
    // max(x,0) - x*y + log1p(exp(-|x|))
    // exp(-|x|) via v_exp_f32: arg <= 0, result in (0,1]  -> no fixup needed
    // log1p(t) = ln2 * v_log_f32(1+t): arg in (1,2]       -> no fixup needed
    const float t = __builtin_amdgcn_exp2f(-LOG2E * fabsf(x));
    const float l = LN2 * __builtin_amdgcn_logf(1.0f + t);
    return fmaxf(x, 0.0f) - x * y + l;
}

template <bool USE_ATOMIC>
__global__ __launch_bounds__(256)
void affinity_wmma_kernel(const float* __restrict__ logits,
                          const int*   __restrict__ labels,
                          float* __restrict__ partials,   // NBLOCKS floats (partials mode)
                          float* __restrict__ out)        // scalar (atomic mode)
{
    const int lane = threadIdx.x & 31;
    const int wid  = threadIdx.x >> 5;
    const int p    = blockIdx.x * WAVES_PER_BLOCK + wid;   // one pixel per wave

    // decompose pixel index -> (n, y0, x0)
    const int n   = p / LPX;
    const int rem = p - n * LPX;
    const int y0  = rem / WP;
    const int x0  = rem - y0 * WP;

    const int  m        = lane & 15;    // A row / D column index (window offset)
    const int  half     = lane >> 4;    // lane half selects K sub-columns
    const bool rowValid = (m < 9);

    // window offset for this lane's row, clamped for padding rows so every
    // address is in-bounds (values for invalid rows are zeroed after load)
    const int j  = rowValid ? m : 0;
    const int dy = j / 3;
    const int dx = j - dy * 3;

    const float* logbase = logits + (size_t)n * CC * PLANE
                                  + (size_t)(y0 + dy) * WW + (x0 + dx);

    // -----------------------------------------------------------------
    // A fragments for V_WMMA_F32_16X16X4_F32 (ISA 7.12.2, 32-bit 16x4 A):
    //   chunk kc, VGPR v, lane L:  A[L%16][4*kc + 2*(L/16) + v]
    // Channels 0..18 real, channel 19 zero-padded (clamped address + cndmask,
    // so all 10 loads are unconditional and issue as one clause).
    // -----------------------------------------------------------------
    v2f a[5];
#pragma unroll
    for (int kc = 0; kc < 5; ++kc) {
        const int c0 = 4 * kc + 2 * half;                   // <= 18 always
        const int c1 = (c0 + 1 < CC) ? (c0 + 1) : (CC - 1); // clamp ch 19 -> 18
        float e0 = logbase[(size_t)c0 * PLANE];
        float e1 = logbase[(size_t)c1 * PLANE];
        e0 = rowValid ? e0 : 0.0f;
        e1 = (rowValid && (c0 + 1) < CC) ? e1 : 0.0f;       // only kc==4,half==1 masks
        a[kc].x = e0;
        a[kc].y = e1;
    }

    // Gram: D = A * A^T, chained over 5 K=4 chunks (B registers == A registers)
    v8f acc = {};
#pragma unroll
    for (int kc = 0; kc < 5; ++kc) {
        acc = __builtin_amdgcn_wmma_f32_16x16x4_f32(
            /*neg_a=*/false, a[kc], /*neg_b=*/false, a[kc],
            /*c_mod=*/(short)0, acc, /*reuse_a=*/false, /*reuse_b=*/false);
    }

    // -----------------------------------------------------------------
    // Labels: 9 window labels with compile-time offsets (vectorizes to wide
    // loads), plus this lane's own column label (avoids a select chain).
    // -----------------------------------------------------------------
    const int* labbase = labels + (size_t)n * PLANE + (size_t)y0 * WW + x0;
    int lab[9];
#pragma unroll
    for (int q = 0; q < 9; ++q)
        lab[q] = labbase[(q / 3) * WW + (q % 3)];
    const int labN = labbase[dy * WW + dx];   // clamped offset: always in-bounds

    // -----------------------------------------------------------------
    // BCE-with-logits on the valid 9x9 sub-block of D.
    // D layout: VGPR r, lane L -> M = r + 8*(L/16), N = L%16.
    // half==0 lanes own M=0..7 (constant lab indices); half==1 lanes own M=8.
    // -----------------------------------------------------------------
    float local = 0.0f;
    if (rowValid) {
        if (half == 0) {
#pragma unroll
            for (int r = 0; r < 8; ++r) {
                const float y = (lab[r] == labN) ? 1.0f : 0.0f;
                local += bce_term(acc[r], y);
            }
        } else {
            const float y = (lab[8] == labN) ? 1.0f : 0.0f;
            local += bce_term(acc[0], y);      // r=0 -> M=8
        }
    }

    // wave32 reduction
#pragma unroll
    for (int off = 16; off > 0; off >>= 1)
        local += __shfl_xor(local, off, 32);

    __shared__ float smem[WAVES_PER_BLOCK];
    if (lane == 0) smem[wid] = local;
    __syncthreads();

    if (threadIdx.x == 0) {
        float s = 0.0f;
#pragma unroll
        for (int i = 0; i < WAVES_PER_BLOCK; ++i) s += smem[i];
        if (USE_ATOMIC) {
            atomicAdd(out, s * INV_TOTAL);
        } else {
            partials[blockIdx.x] = s;
        }
    }
}

__global__ void zero_out_kernel(float* out) { out[0] = 0.0f; }

// deterministic fixed-order final reduction of per-block partials
__global__ __launch_bounds__(256)
void reduce_partials_kernel(const float* __restrict__ partials, int nparts,
                            float* __restrict__ out)
{
    __shared__ float smem[256];
    float s = 0.0f;
    for (int i = threadIdx.x; i < nparts; i += 256)
        s += partials[i];
    smem[threadIdx.x] = s;
    __syncthreads();
#pragma unroll
    for (int stride = 128; stride > 0; stride >>= 1) {
        if (threadIdx.x < stride) smem[threadIdx.x] += smem[threadIdx.x + stride];
        __syncthreads();
    }
    if (threadIdx.x == 0) out[0] = smem[0] * INV_TOTAL;
}

extern "C" void kernel_launch(void* const* d_in, const int* in_sizes, int n_in,
                              void* d_out, int out_size, void* d_ws, size_t ws_size,
                              hipStream_t stream)
{
    const float* logits = (const float*)d_in[0];   // (4, 19, 384, 384) f32
    const int*   labels = (const int*)d_in[1];     // (4, 384, 384) i32
    float*       out    = (float*)d_out;           // scalar f32

    (void)in_sizes; (void)n_in; (void)out_size;

    if (ws_size >= (size_t)NBLOCKS * sizeof(float)) {
        float* partials = (float*)d_ws;
        affinity_wmma_kernel<false><<<NBLOCKS, 256, 0, stream>>>(
            logits, labels, partials, out);
        reduce_partials_kernel<<<1, 256, 0, stream>>>(partials, NBLOCKS, out);
    } else {
        // fallback: float atomics into the output scalar
        zero_out_kernel<<<1, 1, 0, stream>>>(out);
        affinity_wmma_kernel<true><<<NBLOCKS, 256, 0, stream>>>(
            logits, labels, nullptr, out);
    }
}